// _NLblock_68607807586399
// MI455X (gfx1250) — compile-verified
//
#include <hip/hip_runtime.h>
#include <math.h>

typedef __attribute__((ext_vector_type(2))) float v2f;
typedef __attribute__((ext_vector_type(8))) float v8f;

#define WMMA_F32(a,b,c) __builtin_amdgcn_wmma_f32_16x16x4_f32(false,(a),false,(b),(short)0,(c),false,false)

static __device__ __forceinline__ v2f ld2(const float* p) {
  return *reinterpret_cast<const v2f*>(p);
}
static __device__ __forceinline__ v8f zero8() {
  v8f z = {0.f,0.f,0.f,0.f,0.f,0.f,0.f,0.f};
  return z;
}

// ---------------------------------------------------------------------------
// Kernel 1: fused phi/theta/gamma 1x1 convs.
// GEMM per batch: W[768,512] x X[512,9216] -> pg[768,9216], bias + relu(phi,theta).
// Each wave computes a 64(M) x 64(N) tile: 4 A-frags x 4 B-frags -> 16 WMMA
// per K-step (0.75 VMEM instr / WMMA), V_WMMA_F32_16X16X4_F32 throughout.
// ---------------------------------------------------------------------------
__global__ __launch_bounds__(256)
void k1_conv3(const float* __restrict__ x,
              const float* __restrict__ phi_w, const float* __restrict__ phi_b,
              const float* __restrict__ theta_w, const float* __restrict__ theta_b,
              const float* __restrict__ gamma_w, const float* __restrict__ gamma_b,
              float* __restrict__ pg) {
  const int lane = threadIdx.x & 31;
  const int widx = threadIdx.x >> 5;
  const int l16  = lane & 15;
  const int half = lane >> 4;
  const int koff = half * 2;

  int w   = blockIdx.x * 8 + widx;          // 13824 waves total
  int b   = w / (12 * 144);
  int rem = w % (12 * 144);
  int mt  = rem / 144;
  int nt  = rem % 144;
  const int o_base = mt * 64;               // 0..704 (64-row macro tile)
  const int n_base = nt * 64;

  const int rb = o_base >> 8;               // 0=phi 1=theta 2=gamma (64 | 256)
  const float* Wsel = (rb == 0) ? phi_w : (rb == 1 ? theta_w : gamma_w);
  const float* Bsel = (rb == 0) ? phi_b : (rb == 1 ? theta_b : gamma_b);
  const int row_in  = o_base & 255;

  const float* xb    = x + (size_t)b * 512 * 9216;
  const float* wrow0 = Wsel + (size_t)(row_in + l16) * 512;

  v8f acc[4][4];
#pragma unroll
  for (int i = 0; i < 4; ++i)
#pragma unroll
    for (int j = 0; j < 4; ++j) acc[i][j] = zero8();

  for (int k = 0; k < 512; k += 4) {
    v2f a[4];
#pragma unroll
    for (int i = 0; i < 4; ++i)
      a[i] = ld2(wrow0 + (size_t)i * 16 * 512 + k + koff);     // A: 16x4 frags
    const float* xk = xb + (size_t)(k + koff) * 9216 + n_base + l16;
    v2f bf[4];
    bf[0] = v2f{ xk[0],  xk[9216] };                           // B: 4x16 frags
    bf[1] = v2f{ xk[16], xk[16 + 9216] };
    bf[2] = v2f{ xk[32], xk[32 + 9216] };
    bf[3] = v2f{ xk[48], xk[48 + 9216] };
#pragma unroll
    for (int i = 0; i < 4; ++i)
#pragma unroll
      for (int j = 0; j < 4; ++j)
        acc[i][j] = WMMA_F32(a[i], bf[j], acc[i][j]);
  }

  const bool relu = (o_base < 512);         // uniform across the 64-row tile
  float* outb = pg + (size_t)b * 768 * 9216;
#pragma unroll
  for (int i = 0; i < 4; ++i) {
#pragma unroll
    for (int r = 0; r < 8; ++r) {
      const int rl   = i * 16 + half * 8 + r;
      const float bias = Bsel[row_in + rl];
      float* orp = outb + (size_t)(o_base + rl) * 9216 + n_base + l16;
#pragma unroll
      for (int j = 0; j < 4; ++j) {
        float v = acc[i][j][r] + bias;
        if (relu) v = fmaxf(v, 0.f);
        orp[j * 16] = v;
      }
    }
  }
}

// ---------------------------------------------------------------------------
// Kernel 2: SPP pooling for theta (pg rows 256..511) and gamma (512..767).
// Hierarchical: 24x24 grid of 4x4 block sums in LDS, then the 110 pooled means.
// ---------------------------------------------------------------------------
__global__ __launch_bounds__(256)
void k2_spp(const float* __restrict__ pg, float* __restrict__ thp,
            float* __restrict__ gap) {
  __shared__ float S4[576];
  const int idx = blockIdx.x;
  const int b   = idx >> 9;
  const int rem = idx & 511;
  const int sel = rem >> 8;                  // 0=theta 1=gamma
  const int ch  = rem & 255;
  const float* src = pg + (size_t)b * 768 * 9216
                        + (size_t)(256 + (sel << 8) + ch) * 9216;
  const int tid = threadIdx.x;

  for (int cell = tid; cell < 576; cell += 256) {
    const int cy = cell / 24, cx = cell % 24;
    const float* p = src + cy * 4 * 96 + cx * 4;
    float s = 0.f;
#pragma unroll
    for (int r = 0; r < 4; ++r) {
      const float4 v = *reinterpret_cast<const float4*>(p + r * 96);
      s += v.x + v.y + v.z + v.w;
    }
    S4[cell] = s;
  }
  __syncthreads();

  if (tid < 110) {
    float s = 0.f, scale;
    if (tid == 0) {
      for (int i = 0; i < 576; ++i) s += S4[i];
      scale = 1.f / 9216.f;
    } else if (tid < 10) {                   // s=3: 8x8 regions of S4
      const int k = tid - 1, i = k / 3, j = k % 3;
      for (int yy = 0; yy < 8; ++yy)
        for (int xx = 0; xx < 8; ++xx) s += S4[(i * 8 + yy) * 24 + j * 8 + xx];
      scale = 1.f / 1024.f;
    } else if (tid < 46) {                   // s=6: 4x4 regions
      const int k = tid - 10, i = k / 6, j = k % 6;
      for (int yy = 0; yy < 4; ++yy)
        for (int xx = 0; xx < 4; ++xx) s += S4[(i * 4 + yy) * 24 + j * 4 + xx];
      scale = 1.f / 256.f;
    } else {                                 // s=8: 3x3 regions
      const int k = tid - 46, i = k / 8, j = k % 8;
      for (int yy = 0; yy < 3; ++yy)
        for (int xx = 0; xx < 3; ++xx) s += S4[(i * 3 + yy) * 24 + j * 3 + xx];
      scale = 1.f / 144.f;
    }
    float* dst = (sel ? gap : thp) + ((size_t)b * 256 + ch) * 110;
    dst[tid] = s * scale;
  }
}

// ---------------------------------------------------------------------------
// Kernel 3: fused attention. Per 16-row tile:
//   y1 = phi_r[16,256] x theta_p[256,110]  (WMMA, N padded to 7x16)
//   softmax over 110 (in D-layout registers, width-16 shfl reductions)
//   y2 = P[16,112] x gamma_p^T[112,256]    (WMMA, A staged through LDS)
// phi_r rows are contiguous 256-float chunks (raw reshape), y2 stored so the
// later raw reshape back to [256,9216] is free.
// ---------------------------------------------------------------------------
__global__ __launch_bounds__(256)
void k3_attn(const float* __restrict__ pg, const float* __restrict__ thp,
             const float* __restrict__ gap, float* __restrict__ y2) {
  __shared__ float P[8][16][112];            // 57,344 B
  const int lane = threadIdx.x & 31;
  const int widx = threadIdx.x >> 5;
  const int l16  = lane & 15;
  const int half = lane >> 4;
  const int koff = half * 2;

  const int wt = blockIdx.x * 8 + widx;      // 4608 wave-tiles
  const int b  = wt / 576;
  const int m  = wt % 576;
  const int n_base = m * 16;

  const float* phib = pg  + (size_t)b * 768 * 9216;   // phi = first 256 rows
  const float* thb  = thp + (size_t)b * 28160;
  const float* gab  = gap + (size_t)b * 28160;

  // ---- phase 1: y1 = phi_r x theta_p ----
  const float* arow = phib + (size_t)(n_base + l16) * 256;
  v8f acc[7];
#pragma unroll
  for (int t = 0; t < 7; ++t) acc[t] = zero8();

  for (int kc = 0; kc < 256; kc += 4) {
    v2f a = ld2(arow + kc + koff);
    const float* t0 = thb + (size_t)(kc + koff) * 110;
    const float* t1 = t0 + 110;
#pragma unroll
    for (int t = 0; t < 7; ++t) {
      int col = t * 16 + l16; if (col > 109) col = 109;   // clamp, masked later
      v2f bf = { t0[col], t1[col] };
      acc[t] = WMMA_F32(a, bf, acc[t]);
    }
  }

  // ---- softmax over k (110), scale = 9216^-0.25 = 1/sqrt(96) ----
  const float sc = 0.10206207261596575f;
#pragma unroll
  for (int r = 0; r < 8; ++r) {
    float v[7];
    float mx = -3.4e38f;
#pragma unroll
    for (int t = 0; t < 7; ++t) {
      v[t] = acc[t][r] * sc;
      if ((t * 16 + l16) < 110) mx = fmaxf(mx, v[t]);
    }
    for (int s2 = 1; s2 < 16; s2 <<= 1) mx = fmaxf(mx, __shfl_xor(mx, s2, 16));
    float sum = 0.f;
#pragma unroll
    for (int t = 0; t < 7; ++t) {
      v[t] = ((t * 16 + l16) < 110) ? __expf(v[t] - mx) : 0.f;
      sum += v[t];
    }
    for (int s2 = 1; s2 < 16; s2 <<= 1) sum += __shfl_xor(sum, s2, 16);
    const float inv = 1.f / sum;
#pragma unroll
    for (int t = 0; t < 7; ++t) P[widx][half * 8 + r][t * 16 + l16] = v[t] * inv;
  }
  __syncthreads();

  // ---- phase 2: y2 = P x gamma_p^T ----
  v8f acc2[16];
#pragma unroll
  for (int j = 0; j < 16; ++j) acc2[j] = zero8();

  for (int kk = 0; kk < 112; kk += 4) {
    v2f a = { P[widx][l16][kk + koff], P[widx][l16][kk + koff + 1] };
#pragma unroll
    for (int j = 0; j < 16; ++j) {
      const int col = j * 16 + l16;
      v2f bf = ld2(gab + (size_t)col * 110 + kk + koff);  // G[k][c]=gamma_p[c][k]
      acc2[j] = WMMA_F32(a, bf, acc2[j]);
    }
  }

  float* yb = y2 + (size_t)b * 9216 * 256;   // [n][c] contiguous in c
#pragma unroll
  for (int r = 0; r < 8; ++r) {
    const int row = n_base + half * 8 + r;
    float* orp = yb + (size_t)row * 256 + l16;
#pragma unroll
    for (int j = 0; j < 16; ++j) orp[j * 16] = acc2[j][r];
  }
}

// ---------------------------------------------------------------------------
// Kernel 4: out = x + conv1x1(y2_r, W_w, W_b).
// y2 buffer reinterpreted raw as [256][9216] per batch (free reshape).
// GEMM: W_w[512,256] x y2_r[256,9216], 64x64 tile per wave (16 WMMA/K-step).
// ---------------------------------------------------------------------------
__global__ __launch_bounds__(256)
void k4_out(const float* __restrict__ x, const float* __restrict__ Ww,
            const float* __restrict__ Wb, const float* __restrict__ y2,
            float* __restrict__ out) {
  const int lane = threadIdx.x & 31;
  const int widx = threadIdx.x >> 5;
  const int l16  = lane & 15;
  const int half = lane >> 4;
  const int koff = half * 2;

  int w   = blockIdx.x * 8 + widx;           // 9216 waves
  int b   = w / (8 * 144);
  int rem = w % (8 * 144);
  int mt  = rem / 144;
  int nt  = rem % 144;
  const int o_base = mt * 64;
  const int n_base = nt * 64;

  const float* yb    = y2 + (size_t)b * 9216 * 256;   // viewed [256][9216]
  const float* wrow0 = Ww + (size_t)(o_base + l16) * 256;

  v8f acc[4][4];
#pragma unroll
  for (int i = 0; i < 4; ++i)
#pragma unroll
    for (int j = 0; j < 4; ++j) acc[i][j] = zero8();

  for (int k = 0; k < 256; k += 4) {
    v2f a[4];
#pragma unroll
    for (int i = 0; i < 4; ++i)
      a[i] = ld2(wrow0 + (size_t)i * 16 * 256 + k + koff);
    const float* yk = yb + (size_t)(k + koff) * 9216 + n_base + l16;
    v2f bf[4];
    bf[0] = v2f{ yk[0],  yk[9216] };
    bf[1] = v2f{ yk[16], yk[16 + 9216] };
    bf[2] = v2f{ yk[32], yk[32 + 9216] };
    bf[3] = v2f{ yk[48], yk[48 + 9216] };
#pragma unroll
    for (int i = 0; i < 4; ++i)
#pragma unroll
      for (int j = 0; j < 4; ++j)
        acc[i][j] = WMMA_F32(a[i], bf[j], acc[i][j]);
  }

  const float* xb = x   + (size_t)b * 512 * 9216;
  float*       ob = out + (size_t)b * 512 * 9216;
#pragma unroll
  for (int i = 0; i < 4; ++i) {
#pragma unroll
    for (int r = 0; r < 8; ++r) {
      const int row  = o_base + i * 16 + half * 8 + r;
      const float bias = Wb[row];
      const float* xr = xb + (size_t)row * 9216 + n_base + l16;
      float*       orp = ob + (size_t)row * 9216 + n_base + l16;
#pragma unroll
      for (int j = 0; j < 4; ++j)
        orp[j * 16] = acc[i][j][r] + bias + xr[j * 16];
    }
  }
}

// ---------------------------------------------------------------------------
extern "C" void kernel_launch(void* const* d_in, const int* in_sizes, int n_in,
                              void* d_out, int out_size, void* d_ws, size_t ws_size,
                              hipStream_t stream) {
  const float* x       = (const float*)d_in[0];
  const float* phi_w   = (const float*)d_in[1];
  const float* phi_b   = (const float*)d_in[2];
  const float* theta_w = (const float*)d_in[3];
  const float* theta_b = (const float*)d_in[4];
  const float* gamma_w = (const float*)d_in[5];
  const float* gamma_b = (const float*)d_in[6];
  const float* W_w     = (const float*)d_in[7];
  const float* W_b     = (const float*)d_in[8];
  float* out = (float*)d_out;

  char* ws = (char*)d_ws;
  // layout: [theta_p 0..0.9MB][gamma_p @1MB][pg @2MB, 226.5MB][y2, 75.5MB]
  float* thp = (float*)(ws);
  float* gap = (float*)(ws + (size_t)(1u << 20));
  float* pg  = (float*)(ws + (size_t)(2u << 20));
  float* y2  = (float*)(ws + (size_t)(2u << 20) + (size_t)226492416u);

  k1_conv3<<<1728, 256, 0, stream>>>(x, phi_w, phi_b, theta_w, theta_b,
                                     gamma_w, gamma_b, pg);
  k2_spp  <<<4096, 256, 0, stream>>>(pg, thp, gap);
  k3_attn <<<576,  256, 0, stream>>>(pg, thp, gap, y2);
  k4_out  <<<1152, 256, 0, stream>>>(x, W_w, W_b, y2, out);
}